// GumbelSparseAttention_75943611728070
// MI455X (gfx1250) — compile-verified
//
#include <hip/hip_runtime.h>
#include <hip/hip_bf16.h>

typedef __attribute__((ext_vector_type(2))) float v2f;
typedef __attribute__((ext_vector_type(8))) float v8f;

#define EMBED   1024
#define SEQ     1024
#define HEADS   16
#define HD      64
#define BATCH_N 2
#define MROWS   (BATCH_N * SEQ)   // 2048

#define BM 128
#define BN 64
#define BK 32
#define LDS_PAD 36   // 144B row stride: 16B-aligned b128 stores, conflict-free reads

#define WMMA_F32(a, b, c) \
  __builtin_amdgcn_wmma_f32_16x16x4_f32(false, (a), false, (b), (short)0, (c), false, false)

// ---------------------------------------------------------------------------
// C[M,N] = A[M,K] * W[N,K]^T + bias[N]      (all f32, WMMA f32 16x16x4)
// Block: 256 threads (8 wave32). Tile BM=128, BN=64, BK=32.
// Wave w = (wr = w%4, wc = w/4): 2x2 subtiles at rows 32*wr(+16), cols 32*wc(+16).
// 4 WMMAs per 4 LDS fragment loads -> half the DS traffic/waits per flop.
// ---------------------------------------------------------------------------
__global__ __launch_bounds__(256)
void gemm_nt_bias_f32(const float* __restrict__ A, const float* __restrict__ W,
                      const float* __restrict__ bias, float* __restrict__ C,
                      int M, int N, int K) {
  __shared__ alignas(16) float Alds[BM][LDS_PAD];
  __shared__ alignas(16) float Wlds[BN][LDS_PAD];

  const int tid  = threadIdx.x;
  const int wave = tid >> 5;
  const int lane = tid & 31;
  const int bm = blockIdx.x * BM;
  const int bn = blockIdx.y * BN;
  const int m0 = (wave & 3) * 32;
  const int n0 = (wave >> 2) * 32;
  const int ml = lane & 15;           // row (A) / col (B,C) within 16-tile
  const int kk = (lane >> 4) * 2;     // K sub-offset per 16x16x4 layout

  v8f acc00 = {}, acc01 = {}, acc10 = {}, acc11 = {};

  for (int k0 = 0; k0 < K; k0 += BK) {
    // A tile: 128x32 = 1024 float4, 4 per thread; rows of 8 float4
#pragma unroll
    for (int j = 0; j < 4; ++j) {
      const int i = tid + j * 256;
      const int r = i >> 3;
      const int c = (i & 7) * 4;
      const float4 v = *(const float4*)(A + (size_t)(bm + r) * K + k0 + c);
      *(float4*)&Alds[r][c] = v;
    }
    // W tile: 64x32 = 512 float4, 2 per thread
#pragma unroll
    for (int j = 0; j < 2; ++j) {
      const int i = tid + j * 256;
      const int r = i >> 3;
      const int c = (i & 7) * 4;
      const float4 v = *(const float4*)(W + (size_t)(bn + r) * K + k0 + c);
      *(float4*)&Wlds[r][c] = v;
    }
    __syncthreads();

#pragma unroll
    for (int k4 = 0; k4 < BK; k4 += 4) {
      v2f a0, a1, b0, b1;
      a0.x = Alds[m0 + ml][k4 + kk];       a0.y = Alds[m0 + ml][k4 + kk + 1];
      a1.x = Alds[m0 + 16 + ml][k4 + kk];  a1.y = Alds[m0 + 16 + ml][k4 + kk + 1];
      b0.x = Wlds[n0 + ml][k4 + kk];       b0.y = Wlds[n0 + ml][k4 + kk + 1];
      b1.x = Wlds[n0 + 16 + ml][k4 + kk];  b1.y = Wlds[n0 + 16 + ml][k4 + kk + 1];
      acc00 = WMMA_F32(a0, b0, acc00);
      acc01 = WMMA_F32(a0, b1, acc01);
      acc10 = WMMA_F32(a1, b0, acc10);
      acc11 = WMMA_F32(a1, b1, acc11);
    }
    __syncthreads();
  }

  const int colA = bn + n0 + ml;
  const int colB = colA + 16;
  const float bA = bias[colA];
  const float bB = bias[colB];
  const int r0 = bm + m0 + ((lane >> 4) * 8);   // C layout: VGPR r -> M=r / r+8
  const int r1 = r0 + 16;
#pragma unroll
  for (int r = 0; r < 8; ++r) {
    C[(size_t)(r0 + r) * N + colA] = acc00[r] + bA;
    C[(size_t)(r0 + r) * N + colB] = acc01[r] + bB;
    C[(size_t)(r1 + r) * N + colA] = acc10[r] + bA;
    C[(size_t)(r1 + r) * N + colB] = acc11[r] + bB;
  }
}

// ---------------------------------------------------------------------------
// Fused: logits[s,n] = q[b,h,s,:] . Wg[n,:] ;  idx = argmax_n(logits+bg+gumbel)
// Block: 128 threads = 4 waves; each wave owns one 16-query-row tile.
// A fragments live in registers (32 VGPRs); Wg tiles staged in shared LDS.
// ---------------------------------------------------------------------------
__global__ __launch_bounds__(128)
void logits_argmax(const float* __restrict__ Q,      // [B*S, E], head h at col h*64
                   const float* __restrict__ Wg,     // [S, HD]
                   const float* __restrict__ bg,     // [S]
                   const float* __restrict__ gumbel, // [B,H,S,S]
                   int* __restrict__ outIdx) {       // [B*H, S]
  __shared__ alignas(16) float Blds[16][68];

  const int tid  = threadIdx.x;
  const int wave = tid >> 5;
  const int lane = tid & 31;
  const int bh    = blockIdx.x;                 // b*HEADS + h
  const int b     = bh / HEADS;
  const int h     = bh % HEADS;
  const int rowT  = blockIdx.y * 4 + wave;      // 0..63
  const int row0  = rowT * 16;

  const int ml = lane & 15;
  const int kk = (lane >> 4) * 2;

  // preload A fragments for all 16 WMMAs of the K=64 reduction
  const float* qrow = Q + (size_t)(b * SEQ + row0 + ml) * EMBED + h * HD;
  v2f areg[16];
#pragma unroll
  for (int k4 = 0; k4 < 16; ++k4) {
    areg[k4].x = qrow[4 * k4 + kk];
    areg[k4].y = qrow[4 * k4 + kk + 1];
  }

  float best[8]; int bidx[8];
#pragma unroll
  for (int r = 0; r < 8; ++r) { best[r] = -__builtin_inff(); bidx[r] = 0; }

  const float* grow = gumbel + ((size_t)bh * SEQ + row0) * SEQ;

  for (int nt = 0; nt < SEQ / 16; ++nt) {
    __syncthreads();
    // 16x64 Wg tile = 256 float4; 128 threads x 2
#pragma unroll
    for (int it = 0; it < 2; ++it) {
      const int i = tid + it * 128;
      const int r = i >> 4;
      const int c = (i & 15) * 4;
      const float4 wv = *(const float4*)(Wg + (size_t)(nt * 16 + r) * HD + c);
      Blds[r][c + 0] = wv.x; Blds[r][c + 1] = wv.y;
      Blds[r][c + 2] = wv.z; Blds[r][c + 3] = wv.w;
    }
    __syncthreads();

    v8f c = {};
#pragma unroll
    for (int k4 = 0; k4 < 16; ++k4) {
      v2f bf;
      bf.x = Blds[ml][4 * k4 + kk];
      bf.y = Blds[ml][4 * k4 + kk + 1];
      c = WMMA_F32(areg[k4], bf, c);
    }

    const int n = nt * 16 + ml;
    const float bgn = bg[n];
#pragma unroll
    for (int r = 0; r < 8; ++r) {
      const int rr = r + ((lane >> 4) * 8);
      const float val = c[r] + bgn + grow[(size_t)rr * SEQ + n];
      if (val > best[r]) { best[r] = val; bidx[r] = n; }   // first-index ties kept
    }
  }

  // argmax across the 16 lanes holding the same row (xor stays inside halves)
#pragma unroll
  for (int r = 0; r < 8; ++r) {
    float bv = best[r]; int bi = bidx[r];
#pragma unroll
    for (int off = 1; off < 16; off <<= 1) {
      const float ov = __shfl_xor(bv, off, 32);
      const int   oi = __shfl_xor(bi, off, 32);
      if (ov > bv || (ov == bv && oi < bi)) { bv = ov; bi = oi; }
    }
    if ((lane & 15) == 0) {
      const int rr = r + ((lane >> 4) * 8);
      outIdx[(size_t)bh * SEQ + row0 + rr] = bi;
    }
  }
}

// ---------------------------------------------------------------------------
// ctx[b,s, h*64+d] = v[b, idx[b,h,s], h*64+d]   (one-hot attention == gather)
// ---------------------------------------------------------------------------
__global__ __launch_bounds__(256)
void gather_ctx(const float* __restrict__ V, const int* __restrict__ Idx,
                float* __restrict__ Ctx) {
  const int t   = blockIdx.x * blockDim.x + threadIdx.x;   // [B*H*S*16)
  const int q4  = t & 15;
  const int rid = t >> 4;                 // bh*SEQ + s
  const int s   = rid % SEQ;
  const int h   = (rid / SEQ) % HEADS;
  const int b   = rid / (SEQ * HEADS);
  const int src = Idx[rid];
  const float4 val = *(const float4*)(V + (size_t)(b * SEQ + src) * EMBED + h * HD + q4 * 4);
  *(float4*)(Ctx + (size_t)(b * SEQ + s) * EMBED + h * HD + q4 * 4) = val;
}

// ---------------------------------------------------------------------------
extern "C" void kernel_launch(void* const* d_in, const int* in_sizes, int n_in,
                              void* d_out, int out_size, void* d_ws, size_t ws_size,
                              hipStream_t stream) {
  (void)in_sizes; (void)n_in; (void)out_size; (void)ws_size;
  const float* query  = (const float*)d_in[0];
  const float* value  = (const float*)d_in[2];
  const float* Wq     = (const float*)d_in[3];
  const float* bq     = (const float*)d_in[4];
  // d_in[1]=key, d_in[5]=Wk, d_in[6]=bk: numerically dead (probs are one-hot)
  const float* Wv     = (const float*)d_in[7];
  const float* bv     = (const float*)d_in[8];
  const float* Wg     = (const float*)d_in[9];
  const float* bg     = (const float*)d_in[10];
  const float* Wo     = (const float*)d_in[11];
  const float* bo     = (const float*)d_in[12];
  const float* gumbel = (const float*)d_in[13];

  char*  ws   = (char*)d_ws;
  float* qbuf = (float*)(ws);                       //  8 MB  [2048,1024]
  float* vbuf = (float*)(ws + ((size_t)8  << 20));  //  8 MB
  float* ctx  = (float*)(ws + ((size_t)16 << 20));  //  8 MB
  int*   idx  = (int*)  (ws + ((size_t)24 << 20));  // 128 KB [32,1024]

  const dim3 gblk(256);
  const dim3 ggrid(MROWS / BM, EMBED / BN);         // 16 x 16

  hipLaunchKernelGGL(gemm_nt_bias_f32, ggrid, gblk, 0, stream,
                     query, Wq, bq, qbuf, MROWS, EMBED, EMBED);
  hipLaunchKernelGGL(gemm_nt_bias_f32, ggrid, gblk, 0, stream,
                     value, Wv, bv, vbuf, MROWS, EMBED, EMBED);

  const dim3 lgrid(BATCH_N * HEADS, SEQ / 64);      // 32 x 16 blocks of 128
  hipLaunchKernelGGL(logits_argmax, lgrid, dim3(128), 0, stream,
                     qbuf, Wg, bg, gumbel, idx);

  const int nthr = BATCH_N * HEADS * SEQ * 16;      // 512K float4 lanes
  hipLaunchKernelGGL(gather_ctx, dim3(nthr / 256), dim3(256), 0, stream,
                     vbuf, idx, ctx);

  hipLaunchKernelGGL(gemm_nt_bias_f32, ggrid, gblk, 0, stream,
                     ctx, Wo, bo, (float*)d_out, MROWS, EMBED, EMBED);
}